// TorchIEKF_38345468019165
// MI455X (gfx1250) — compile-verified
//
#include <hip/hip_runtime.h>
#include <math.h>

typedef __attribute__((ext_vector_type(2))) float v2f;
typedef __attribute__((ext_vector_type(8))) float v8f;

#define DEV static __device__ __forceinline__

// ---------------- small 3x3 helpers (thread-0 scalar work) ----------------
DEV void m3mul(const float* A, const float* B, float* C) {
#pragma unroll
  for (int i = 0; i < 3; ++i)
#pragma unroll
    for (int j = 0; j < 3; ++j)
      C[i*3+j] = A[i*3+0]*B[0+j] + A[i*3+1]*B[3+j] + A[i*3+2]*B[6+j];
}
DEV void m3vec(const float* A, const float* v, float* o) {
#pragma unroll
  for (int i = 0; i < 3; ++i)
    o[i] = A[i*3+0]*v[0] + A[i*3+1]*v[1] + A[i*3+2]*v[2];
}
DEV void m3Tvec(const float* A, const float* v, float* o) {
#pragma unroll
  for (int i = 0; i < 3; ++i)
    o[i] = A[0*3+i]*v[0] + A[1*3+i]*v[1] + A[2*3+i]*v[2];
}
DEV void skew3(const float* v, float* S) {
  S[0] = 0.f;    S[1] = -v[2];  S[2] = v[1];
  S[3] = v[2];   S[4] = 0.f;    S[5] = -v[0];
  S[6] = -v[1];  S[7] = v[0];   S[8] = 0.f;
}
DEV void so3exp(const float* phi, float* R) {
  float a2 = phi[0]*phi[0] + phi[1]*phi[1] + phi[2]*phi[2];
  if (a2 < 1e-12f) {
    float S[9]; skew3(phi, S);
#pragma unroll
    for (int k = 0; k < 9; ++k) R[k] = ((k % 4) == 0 ? 1.f : 0.f) + S[k];
  } else {
    float a = sqrtf(a2), inva = 1.f / a;
    float ax = phi[0]*inva, ay = phi[1]*inva, az = phi[2]*inva;
    float c = cosf(a), s = sinf(a), mc = 1.f - c;
    R[0] = c + mc*ax*ax;      R[1] = mc*ax*ay - s*az;  R[2] = mc*ax*az + s*ay;
    R[3] = mc*ax*ay + s*az;   R[4] = c + mc*ay*ay;     R[5] = mc*ay*az - s*ax;
    R[6] = mc*ax*az - s*ay;   R[7] = mc*ay*az + s*ax;  R[8] = c + mc*az*az;
  }
}
DEV void sen3exp9(const float* xi, float* R, float* x0, float* x1) {
  float a2 = xi[0]*xi[0] + xi[1]*xi[1] + xi[2]*xi[2];
  float J[9];
  if (a2 < 1e-12f) {
    float S[9]; skew3(xi, S);
#pragma unroll
    for (int k = 0; k < 9; ++k) {
      float id = ((k % 4) == 0 ? 1.f : 0.f);
      R[k] = id + S[k];
      J[k] = id + 0.5f * S[k];
    }
  } else {
    float a = sqrtf(a2), inva = 1.f / a;
    float ax = xi[0]*inva, ay = xi[1]*inva, az = xi[2]*inva;
    float c = cosf(a), s = sinf(a), mc = 1.f - c;
    R[0] = c + mc*ax*ax;      R[1] = mc*ax*ay - s*az;  R[2] = mc*ax*az + s*ay;
    R[3] = mc*ax*ay + s*az;   R[4] = c + mc*ay*ay;     R[5] = mc*ay*az - s*ax;
    R[6] = mc*ax*az - s*ay;   R[7] = mc*ay*az + s*ax;  R[8] = c + mc*az*az;
    float sa = s * inva, msa = 1.f - sa, ca = mc * inva;
    J[0] = sa + msa*ax*ax;     J[1] = msa*ax*ay - ca*az; J[2] = msa*ax*az + ca*ay;
    J[3] = msa*ax*ay + ca*az;  J[4] = sa + msa*ay*ay;    J[5] = msa*ay*az - ca*ax;
    J[6] = msa*ax*az - ca*ay;  J[7] = msa*ay*az + ca*ax; J[8] = sa + msa*az*az;
  }
  m3vec(J, xi + 3, x0);
  m3vec(J, xi + 6, x1);
}
DEV void normalize_rot3(float* R) {
#pragma unroll 1
  for (int it = 0; it < 8; ++it) {
    float A[9];
#pragma unroll
    for (int i = 0; i < 3; ++i)
#pragma unroll
      for (int j = 0; j < 3; ++j)
        A[i*3+j] = R[i*3+0]*R[j*3+0] + R[i*3+1]*R[j*3+1] + R[i*3+2]*R[j*3+2];
    float M[9]; m3mul(A, R, M);
#pragma unroll
    for (int k = 0; k < 9; ++k) R[k] = 1.5f * R[k] - 0.5f * M[k];
  }
}
DEV void from_rpy3(float r, float p, float y, float* R) {
  float cr = cosf(r), sr = sinf(r), cp = cosf(p), sp = sinf(p), cy = cosf(y), sy = sinf(y);
  float Rx[9] = {1,0,0, 0,cr,-sr, 0,sr,cr};
  float Ry[9] = {cp,0,sp, 0,1,0, -sp,0,cp};
  float Rz[9] = {cy,-sy,0, sy,cy,0, 0,0,1};
  float T[9]; m3mul(Rz, Ry, T); m3mul(T, Rx, R);
}

// ------------- 32x32 f32 matmul on the matrix engine (4 waves) -------------
// C = A @ B (or A @ B^T), all row-major 32x32 in LDS, zero-padded beyond 21.
// Wave w owns output tile (w>>1, w&1); 8 x V_WMMA_F32_16X16X4_F32 per wave.
DEV void mm32(const float* A, const float* B, float* C, bool transB, int tid) {
  const int wave  = tid >> 5;
  const int lane  = tid & 31;
  const int tm    = (wave >> 1) << 4;   // tile row base
  const int tn    = (wave & 1) << 4;    // tile col base
  const int r16   = lane & 15;          // m for A-frag, n for B/C-frag
  const int khalf = (lane >> 4) << 1;   // 0 or 2
  v8f acc = {};
#pragma unroll
  for (int k0 = 0; k0 < 32; k0 += 4) {
    v2f a, b;
    a.x = A[(tm + r16) * 32 + (k0 + khalf)];
    a.y = A[(tm + r16) * 32 + (k0 + khalf + 1)];
    if (!transB) {
      b.x = B[(k0 + khalf) * 32 + (tn + r16)];
      b.y = B[(k0 + khalf + 1) * 32 + (tn + r16)];
    } else {
      b.x = B[(tn + r16) * 32 + (k0 + khalf)];
      b.y = B[(tn + r16) * 32 + (k0 + khalf + 1)];
    }
    acc = __builtin_amdgcn_wmma_f32_16x16x4_f32(false, a, false, b,
                                                (short)0, acc, false, false);
  }
  const int mhalf = (lane >> 4) << 3;   // 0 or 8
#pragma unroll
  for (int i = 0; i < 8; ++i)
    C[(tm + mhalf + i) * 32 + (tn + r16)] = acc[i];
}

// --------------------------------- kernel ---------------------------------
__global__ void __launch_bounds__(128, 1)
iekf_kernel(const float* __restrict__ t, const float* __restrict__ u,
            const float* __restrict__ mcov, const float* __restrict__ ang0,
            const float* __restrict__ v_mes, float* __restrict__ out, int N) {
  __shared__ float sP[1024], sA[1024], sB[1024], sC[1024], sD[1024];
  __shared__ float sRot[9], sRci[9], sRotN[9];
  __shared__ float sV[3], sPos[3], sBom[3], sBac[3], sTci[3], sVN[3], sPosN[3];
  __shared__ float sH0[21], sH1[21], sHP0[21], sHP1[21], sKT0[21], sKT1[21];
  __shared__ float sRRt[9], sScal[8];

  const int tid = threadIdx.x;
  const size_t Ns = (size_t)N;

  // ---- init: carry0 + outputs[0] ----
  for (int e = tid; e < 1024; e += 128) sP[e] = 0.f;
  __syncthreads();
  if (tid == 0) {
    sP[0*32+0] = 0.001f; sP[1*32+1] = 0.001f;
    sP[3*32+3] = 0.1f;   sP[4*32+4] = 0.1f;
    for (int k = 9;  k < 12; ++k) sP[k*33] = 0.006f;
    for (int k = 12; k < 15; ++k) sP[k*33] = 0.004f;
    for (int k = 15; k < 18; ++k) sP[k*33] = 1e-6f;
    for (int k = 18; k < 21; ++k) sP[k*33] = 0.005f;
    float R0[9]; from_rpy3(ang0[0], ang0[1], ang0[2], R0);
    for (int k = 0; k < 9; ++k) {
      sRot[k] = R0[k];
      sRci[k] = ((k % 4) == 0) ? 1.f : 0.f;
      out[k] = R0[k];                       // Rot[0]
      out[21*Ns + k] = sRci[k];             // Rot_c_i[0]
    }
    for (int k = 0; k < 3; ++k) {
      sV[k] = v_mes[k]; sPos[k] = 0.f; sBom[k] = 0.f; sBac[k] = 0.f; sTci[k] = 0.f;
      out[ 9*Ns + k] = sV[k];               // v[0]
      out[12*Ns + k] = 0.f;                 // p[0]
      out[15*Ns + k] = 0.f;                 // b_omega[0]
      out[18*Ns + k] = 0.f;                 // b_acc[0]
      out[30*Ns + k] = 0.f;                 // t_c_i[0]
    }
  }
  __syncthreads();

  for (int i = 1; i < N; ++i) {
    // ---- phase 1: zero F*dt buffer ----
    for (int e = tid; e < 1024; e += 128) sA[e] = 0.f;
    __syncthreads();

    // ---- phase 2: scalar state propagation, F fill, H fill (thread 0) ----
    if (tid == 0) {
      float dt = t[i] - t[i-1], dt2 = dt * dt;
      float uu[6];
      for (int k = 0; k < 6; ++k) uu[k] = u[(size_t)i*6 + k];
      float Rot[9], Rci[9];
      for (int k = 0; k < 9; ++k) { Rot[k] = sRot[k]; Rci[k] = sRci[k]; }
      float om[3], am[3];
      for (int k = 0; k < 3; ++k) { om[k] = uu[k] - sBom[k]; am[k] = uu[3+k] - sBac[k]; }
      float acc[3]; m3vec(Rot, am, acc); acc[2] -= 9.80665f;
      float vN[3], pN[3];
      for (int k = 0; k < 3; ++k) {
        vN[k] = sV[k] + acc[k]*dt;
        pN[k] = sPos[k] + sV[k]*dt + 0.5f*acc[k]*dt2;
      }
      float phi[3] = {om[0]*dt, om[1]*dt, om[2]*dt};
      float dRm[9]; so3exp(phi, dRm);
      float RotN[9]; m3mul(Rot, dRm, RotN);

      // F*dt (uses OLD Rot, v, p)
      sA[3*32+1] =  9.80665f*dt;  sA[4*32+0] = -9.80665f*dt;
      sA[6*32+3] = dt; sA[7*32+4] = dt; sA[8*32+5] = dt;
      float vOld[3] = {sV[0], sV[1], sV[2]}, pOld[3] = {sPos[0], sPos[1], sPos[2]};
      float SK[9], T[9];
      skew3(vOld, SK); m3mul(SK, Rot, T);
      for (int r = 0; r < 3; ++r) for (int c = 0; c < 3; ++c)
        sA[(3+r)*32 + 9 + c] = -T[r*3+c]*dt;
      skew3(pOld, SK); m3mul(SK, Rot, T);
      for (int r = 0; r < 3; ++r) for (int c = 0; c < 3; ++c)
        sA[(6+r)*32 + 9 + c] = -T[r*3+c]*dt;
      for (int r = 0; r < 3; ++r) for (int c = 0; c < 3; ++c) {
        float m = -Rot[r*3+c]*dt;
        sA[r*32 + 9 + c] = m;        // F[:3,9:12]  = -Rot
        sA[(3+r)*32 + 12 + c] = m;   // F[3:6,12:15]= -Rot
      }
      for (int r = 0; r < 3; ++r) for (int c = 0; c < 3; ++c)
        sRRt[r*3+c] = Rot[r*3]*Rot[c*3] + Rot[r*3+1]*Rot[c*3+1] + Rot[r*3+2]*Rot[c*3+2];

      // measurement model (uses NEW Rot_n, v_n)
      float RotBody[9]; m3mul(RotN, Rci, RotBody);
      float vImu[3];  m3Tvec(RotN, vN, vImu);
      float tmp3[3];  m3Tvec(Rci, vImu, tmp3);
      float SKt[9];   skew3(sTci, SKt);
      float vBody[3];
      for (int k = 0; k < 3; ++k)
        vBody[k] = tmp3[k] + SKt[k*3+0]*om[0] + SKt[k*3+1]*om[1] + SKt[k*3+2]*om[2];
      for (int k = 0; k < 21; ++k) { sH0[k] = 0.f; sH1[k] = 0.f; }
      for (int j = 0; j < 3; ++j) { sH0[3+j] = RotBody[j*3+1]; sH1[3+j] = RotBody[j*3+2]; }
      float SKv[9]; skew3(vImu, SKv);
      float Mrc[9];
      for (int r = 0; r < 3; ++r) for (int c = 0; c < 3; ++c)
        Mrc[r*3+c] = Rci[0*3+r]*SKv[0*3+c] + Rci[1*3+r]*SKv[1*3+c] + Rci[2*3+r]*SKv[2*3+c];
      for (int j = 0; j < 3; ++j) { sH0[15+j] = Mrc[3+j]; sH1[15+j] = Mrc[6+j]; }
      for (int j = 0; j < 3; ++j) { sH0[9+j]  = SKt[3+j]; sH1[9+j]  = SKt[6+j]; }
      float SKo[9]; skew3(om, SKo);
      for (int j = 0; j < 3; ++j) { sH0[18+j] = -SKo[3+j]; sH1[18+j] = -SKo[6+j]; }

      sScal[0] = dt; sScal[1] = dt2;
      sScal[2] = mcov[(size_t)i*2 + 0]; sScal[3] = mcov[(size_t)i*2 + 1];
      sScal[4] = -vBody[1]; sScal[5] = -vBody[2];
      for (int k = 0; k < 9; ++k) sRotN[k] = RotN[k];
      for (int k = 0; k < 3; ++k) { sVN[k] = vN[k]; sPosN[k] = pN[k]; }
    }
    __syncthreads();

    // ---- phase 3: F2 = F@F, F3 = F2@F  (WMMA) ----
    mm32(sA, sA, sB, false, tid); __syncthreads();
    mm32(sB, sA, sC, false, tid); __syncthreads();

    // ---- phase 4: Phi = I + F + F2/2 + F3/6 ; M = P + GQG^T ----
    for (int e = tid; e < 1024; e += 128) {
      int ii = e >> 5, jj = e & 31;
      float idv = (ii == jj && ii < 21) ? 1.f : 0.f;
      sD[e] = idv + sA[e] + 0.5f*sB[e] + (1.f/6.f)*sC[e];
    }
    __syncthreads();
    for (int e = tid; e < 1024; e += 128) sA[e] = sP[e];
    __syncthreads();
    if (tid == 0) {
      float dt2 = sScal[1];
      float c0 = 0.001f*dt2, c1 = 0.01f*dt2;
      for (int r = 0; r < 3; ++r) for (int c = 0; c < 3; ++c) {
        sA[r*32 + c]         += c0 * sRRt[r*3+c];
        sA[(3+r)*32 + 3 + c] += c1 * sRRt[r*3+c];
      }
      for (int k = 9;  k < 12; ++k) sA[k*33] += 6e-9f  * dt2;
      for (int k = 12; k < 15; ++k) sA[k*33] += 2e-4f  * dt2;
      for (int k = 15; k < 18; ++k) sA[k*33] += 1e-9f  * dt2;
      for (int k = 18; k < 21; ++k) sA[k*33] += 1e-9f  * dt2;
    }
    __syncthreads();

    // ---- phase 5: Pn = Phi @ M @ Phi^T  (WMMA) ----
    mm32(sD, sA, sB, false, tid); __syncthreads();   // T1 = Phi@M
    mm32(sB, sD, sC, true,  tid); __syncthreads();   // Pn = T1@Phi^T -> sC

    // ---- phase 6: HP = H @ Pn (2x21) ----
    if (tid < 42) {
      int a = tid & 1, col = tid >> 1;
      const float* H = a ? sH1 : sH0;
      float s = 0.f;
      for (int j = 0; j < 21; ++j) s += H[j] * sC[j*32 + col];
      (a ? sHP1 : sHP0)[col] = s;
    }
    __syncthreads();

    // ---- phase 7: gain, dx, state update, outputs (thread 0) ----
    if (tid == 0) {
      float Rm0 = sScal[2], Rm1 = sScal[3], r0 = sScal[4], r1 = sScal[5];
      float S00 = Rm0, S01 = 0.f, S10 = 0.f, S11 = Rm1;
      for (int j = 0; j < 21; ++j) {
        S00 += sHP0[j]*sH0[j]; S01 += sHP0[j]*sH1[j];
        S10 += sHP1[j]*sH0[j]; S11 += sHP1[j]*sH1[j];
      }
      float inv = 1.f / (S00*S11 - S01*S10);
      float i00 = S11*inv, i01 = -S01*inv, i10 = -S10*inv, i11 = S00*inv;
      float dx[21];
      for (int j = 0; j < 21; ++j) {
        float k0 = i00*sHP0[j] + i01*sHP1[j];
        float k1 = i10*sHP0[j] + i11*sHP1[j];
        sKT0[j] = k0; sKT1[j] = k1;
        dx[j] = k0*r0 + k1*r1;
      }
      float dR[9], x0[3], x1[3];
      sen3exp9(dx, dR, x0, x1);
      float RotN[9]; for (int k = 0; k < 9; ++k) RotN[k] = sRotN[k];
      float RotU[9]; m3mul(dR, RotN, RotU);
      float vNl[3] = {sVN[0], sVN[1], sVN[2]}, pNl[3] = {sPosN[0], sPosN[1], sPosN[2]};
      float vU[3], pU[3];
      m3vec(dR, vNl, vU); m3vec(dR, pNl, pU);
      for (int k = 0; k < 3; ++k) { vU[k] += x0[k]; pU[k] += x1[k]; }
      float bomU[3], bacU[3], tciU[3];
      for (int k = 0; k < 3; ++k) {
        bomU[k] = sBom[k] + dx[9+k];
        bacU[k] = sBac[k] + dx[12+k];
        tciU[k] = sTci[k] + dx[18+k];
      }
      float dRc[9]; so3exp(dx + 15, dRc);
      float Rcil[9]; for (int k = 0; k < 9; ++k) Rcil[k] = sRci[k];
      float RciU[9]; m3mul(dRc, Rcil, RciU);
      if ((i % 100)  == 0) normalize_rot3(RotU);
      if ((i % 1000) == 0) normalize_rot3(RciU);
      for (int k = 0; k < 9; ++k) {
        sRot[k] = RotU[k]; sRci[k] = RciU[k];
        out[(size_t)i*9 + k] = RotU[k];
        out[21*Ns + (size_t)i*9 + k] = RciU[k];
      }
      for (int k = 0; k < 3; ++k) {
        sV[k] = vU[k]; sPos[k] = pU[k]; sBom[k] = bomU[k]; sBac[k] = bacU[k]; sTci[k] = tciU[k];
        out[ 9*Ns + (size_t)i*3 + k] = vU[k];
        out[12*Ns + (size_t)i*3 + k] = pU[k];
        out[15*Ns + (size_t)i*3 + k] = bomU[k];
        out[18*Ns + (size_t)i*3 + k] = bacU[k];
        out[30*Ns + (size_t)i*3 + k] = tciU[k];
      }
    }
    __syncthreads();

    // ---- phase 8: IKH = I - K@H ----
    for (int e = tid; e < 1024; e += 128) {
      int ii = e >> 5, jj = e & 31;
      float v = 0.f;
      if (ii < 21 && jj < 21)
        v = (ii == jj ? 1.f : 0.f) - sKT0[ii]*sH0[jj] - sKT1[ii]*sH1[jj];
      sA[e] = v;
    }
    __syncthreads();

    // ---- phase 9: E = IKH @ Pn @ IKH^T  (WMMA) ----
    mm32(sA, sC, sD, false, tid); __syncthreads();   // T2 = IKH@Pn
    mm32(sD, sA, sB, true,  tid); __syncthreads();   // E  = T2@IKH^T -> sB

    // ---- phase 10: P = sym(E) + K Rm K^T ----
    for (int e = tid; e < 1024; e += 128) {
      int ii = e >> 5, jj = e & 31;
      float pv = 0.f;
      if (ii < 21 && jj < 21)
        pv = 0.5f*(sB[ii*32+jj] + sB[jj*32+ii])
           + sScal[2]*sKT0[ii]*sKT0[jj] + sScal[3]*sKT1[ii]*sKT1[jj];
      sP[e] = pv;
    }
    __syncthreads();
  }
}

extern "C" void kernel_launch(void* const* d_in, const int* in_sizes, int n_in,
                              void* d_out, int out_size, void* d_ws, size_t ws_size,
                              hipStream_t stream) {
  const float* t     = (const float*)d_in[0];
  const float* u     = (const float*)d_in[1];
  const float* mcov  = (const float*)d_in[2];
  const float* v_mes = (const float*)d_in[3];
  // d_in[4] = p_mes (unused by the reference forward pass)
  const float* ang0  = (const float*)d_in[5];
  int N = in_sizes[0];
  float* out = (float*)d_out;
  hipLaunchKernelGGL(iekf_kernel, dim3(1), dim3(128), 0, stream,
                     t, u, mcov, ang0, v_mes, out, N);
}